// GCN_Module_35527969473092
// MI455X (gfx1250) — compile-verified
//
#include <hip/hip_runtime.h>
#include <math.h>

#define BB 512
#define NN 128
#define NFG 8
#define NFR 256
#define NG 4

typedef float v2f __attribute__((ext_vector_type(2)));
typedef float v8f __attribute__((ext_vector_type(8)));

// ---------------------------------------------------------------------------
// Precompute per-head reduced operators:
//   M[g][i][j] = sum_r Wt[g,i,r] * Wp[g,j,r]      (ws[g*64 + i*8 + j])
//   u[g][i]    = sum_r Wt[g,i,r] * bp[g,r]        (ws[256 + g*8 + i])
//   v[g][j]    = sum_r Wp[g,j,r] * bt[g,r]        (ws[288 + g*8 + j])
//   c[g]       = sum_r bt[g,r]  * bp[g,r]         (ws[320 + g])
// This collapses the NFR=256 dimension out of the whole problem.
// ---------------------------------------------------------------------------
__global__ void precompute_heads(const float* __restrict__ Wt,
                                 const float* __restrict__ bt,
                                 const float* __restrict__ Wp,
                                 const float* __restrict__ bp,
                                 float* __restrict__ ws) {
  const int g = blockIdx.x;
  const int t = threadIdx.x;           // 0..63
  const int i = t >> 3, j = t & 7;
  const float* wt  = Wt + (size_t)g * NFG * NFR;
  const float* wp  = Wp + (size_t)g * NFG * NFR;
  const float* btg = bt + g * NFR;
  const float* bpg = bp + g * NFR;

  float m = 0.f;
  for (int r = 0; r < NFR; ++r) m += wt[i * NFR + r] * wp[j * NFR + r];
  ws[g * 64 + t] = m;

  if (t < NFG) {
    float uu = 0.f, vv = 0.f;
    for (int r = 0; r < NFR; ++r) {
      uu += wt[t * NFR + r] * bpg[r];
      vv += wp[t * NFR + r] * btg[r];
    }
    ws[256 + g * NFG + t] = uu;
    ws[288 + g * NFG + t] = vv;
  }
  if (t == 0) {
    float cc = 0.f;
    for (int r = 0; r < NFR; ++r) cc += btg[r] * bpg[r];
    ws[320 + g] = cc;
  }
}

// ---------------------------------------------------------------------------
// Fused per-batch kernel: 256 threads = 8 waves, one block per b.
// Wave w owns sim rows [16w, 16w+16). sim tile = Y(16x8) @ X'^T(8x16) via two
// V_WMMA_F32_16X16X4_F32 steps (K=0..3, K=4..7).
// ---------------------------------------------------------------------------
__global__ __launch_bounds__(256)
void gcn_fused(const float* __restrict__ x_in,
               const float* __restrict__ boxes,
               const float* __restrict__ We,
               const float* __restrict__ be,
               const float* __restrict__ Wg,
               const int*   __restrict__ gtype,
               const float* __restrict__ hw,        // M/u/v/c from d_ws
               float* __restrict__ out_feats,       // [B,N,NFG]
               float* __restrict__ out_attn) {      // [B,N,N] (head NG-1)
  __shared__ float xp[NN][NFG];      // x' = relu(x@We+be)
  __shared__ float px[NN], py[NN];   // box centers
  __shared__ float yv[NN][NFG];      // y = x' @ M[g]
  __shared__ float aArr[NN];         // x'_n . u[g]
  __shared__ float bArr[NN];         // x'_m . v[g]
  __shared__ float facc[NN][NFG];    // sum over heads of relu(agg@Wg)
  __shared__ float wg[NFG][NFG];

  const int b    = blockIdx.x;
  const int tid  = threadIdx.x;
  const int lane = tid & 31;
  const int wave = tid >> 5;         // 0..7
  const int half = lane >> 4;        // 0/1
  const int lm   = lane & 15;
  const int rowBase = wave * 16;

  // ---- per-block setup: positions, x', feats accumulator ----
  for (int n = tid; n < NN; n += 256) {
    px[n] = boxes[((size_t)b * NN + n) * 4 + 0];
    py[n] = boxes[((size_t)b * NN + n) * 4 + 1];
  }
  for (int e = tid; e < NN * NFG; e += 256) {
    const int n = e >> 3, o = e & 7;
    float s = be[o];
#pragma unroll
    for (int i = 0; i < NFG; ++i)
      s += x_in[((size_t)b * NN + n) * NFG + i] * We[i * NFG + o];
    xp[n][o]   = fmaxf(s, 0.f);
    facc[n][o] = 0.f;
  }
  const float thr2 = (gtype[0] == 0) ? 100.0f : 16.0f;  // thr^2 (10 or 4)
  __syncthreads();

  for (int g = 0; g < NG; ++g) {
    // ---- phase A: y = x'@M[g], a_n, b_m, Wg[g] into LDS ----
    for (int e = tid; e < NN * NFG; e += 256) {
      const int n = e >> 3, o = e & 7;
      float s = 0.f;
#pragma unroll
      for (int i = 0; i < NFG; ++i) s += xp[n][i] * hw[g * 64 + i * 8 + o];
      yv[n][o] = s;
    }
    if (tid < NN) {
      float av = 0.f, bv = 0.f;
#pragma unroll
      for (int i = 0; i < NFG; ++i) {
        av += xp[tid][i] * hw[256 + g * NFG + i];
        bv += xp[tid][i] * hw[288 + g * NFG + i];
      }
      aArr[tid] = av;
      bArr[tid] = bv;
    }
    if (tid < NFG * NFG) wg[tid >> 3][tid & 7] = Wg[g * 64 + tid];
    __syncthreads();
    const float cg = hw[320 + g];

    // ---- phase B: WMMA sim tiles + softmax + agg (per wave) ----
    // A fragment: 16x4 f32 layout -> V0: K0(lo)/K2(hi), V1: K1(lo)/K3(hi)
    const int arow = rowBase + lm;
    const int kb   = half * 2;
    v2f a0, a1;
    a0.x = yv[arow][kb + 0]; a0.y = yv[arow][kb + 1];  // K-step 0 (K=0..3)
    a1.x = yv[arow][kb + 4]; a1.y = yv[arow][kb + 5];  // K-step 1 (K=4..7)

    // j-indexed row constants (C/D layout: VGPR j -> row j + 8*half)
    float a_r[8], px_r[8], py_r[8];
#pragma unroll
    for (int j = 0; j < 8; ++j) {
      const int row = rowBase + j + half * 8;
      a_r[j] = aArr[row]; px_r[j] = px[row]; py_r[j] = py[row];
    }

    float sim[8][8];  // [tile t][accum reg j]
#pragma unroll
    for (int t = 0; t < 8; ++t) {
      const int col = t * 16 + lm;
      // B fragment: x'^T 4x16, same K split as A
      v2f b0, b1;
      b0.x = xp[col][kb + 0]; b0.y = xp[col][kb + 1];
      b1.x = xp[col][kb + 4]; b1.y = xp[col][kb + 5];

      v8f acc = {0.f, 0.f, 0.f, 0.f, 0.f, 0.f, 0.f, 0.f};
      acc = __builtin_amdgcn_wmma_f32_16x16x4_f32(false, a0, false, b0,
                                                  (short)0, acc, false, false);
      acc = __builtin_amdgcn_wmma_f32_16x16x4_f32(false, a1, false, b1,
                                                  (short)0, acc, false, false);

      const float bcol = bArr[col];
      const float pxc = px[col], pyc = py[col];
#pragma unroll
      for (int j = 0; j < 8; ++j) {
        const float dx = px_r[j] - pxc, dy = py_r[j] - pyc;
        const float logit = (acc[j] + a_r[j] + bcol + cg) * 0.0625f; // /sqrt(256)
        sim[t][j] = (dx * dx + dy * dy > thr2) ? -INFINITY : logit;
      }
    }

    // ---- softmax over m (8 tiles x 16 lanes of this half) ----
#pragma unroll
    for (int j = 0; j < 8; ++j) {
      float rm = -INFINITY;
#pragma unroll
      for (int t = 0; t < 8; ++t) rm = fmaxf(rm, sim[t][j]);
#pragma unroll
      for (int off = 1; off < 16; off <<= 1)
        rm = fmaxf(rm, __shfl_xor(rm, off, 32));
      float s = 0.f;
#pragma unroll
      for (int t = 0; t < 8; ++t) {
        const float e = __expf(sim[t][j] - rm);
        sim[t][j] = e; s += e;
      }
#pragma unroll
      for (int off = 1; off < 16; off <<= 1) s += __shfl_xor(s, off, 32);
      const float inv = 1.0f / s;
#pragma unroll
      for (int t = 0; t < 8; ++t) sim[t][j] *= inv;
    }

    // ---- stream attn of the last head to HBM (write-once, nontemporal) ----
    if (g == NG - 1) {
#pragma unroll
      for (int t = 0; t < 8; ++t)
#pragma unroll
        for (int j = 0; j < 8; ++j) {
          const int row = rowBase + j + half * 8;
          const int col = t * 16 + lm;
          __builtin_nontemporal_store(
              sim[t][j], &out_attn[((size_t)b * NN + row) * NN + col]);
        }
    }

    // ---- agg = attn @ x' (K=128) : per-lane FMA + 16-lane shuffle reduce ----
    float pagg[8][8];  // [j][f]
#pragma unroll
    for (int j = 0; j < 8; ++j)
#pragma unroll
      for (int f = 0; f < 8; ++f) pagg[j][f] = 0.f;
#pragma unroll
    for (int t = 0; t < 8; ++t) {
      const int col = t * 16 + lm;
      float xc[8];
#pragma unroll
      for (int f = 0; f < 8; ++f) xc[f] = xp[col][f];
#pragma unroll
      for (int j = 0; j < 8; ++j)
#pragma unroll
        for (int f = 0; f < 8; ++f) pagg[j][f] += sim[t][j] * xc[f];
    }
#pragma unroll
    for (int off = 1; off < 16; off <<= 1)
#pragma unroll
      for (int j = 0; j < 8; ++j)
#pragma unroll
        for (int f = 0; f < 8; ++f)
          pagg[j][f] += __shfl_xor(pagg[j][f], off, 32);

    // ---- head_out = relu(agg @ Wg[g]); accumulate into feats ----
    if (lm == 0) {  // lanes 0 and 16: disjoint 8-row groups per wave
#pragma unroll
      for (int j = 0; j < 8; ++j) {
        const int row = rowBase + j + half * 8;
#pragma unroll
        for (int o = 0; o < 8; ++o) {
          float s = 0.f;
#pragma unroll
          for (int f = 0; f < 8; ++f) s += pagg[j][f] * wg[f][o];
          facc[row][o] += fmaxf(s, 0.f);
        }
      }
    }
    __syncthreads();  // protect yv/aArr/bArr/wg/facc for next head
  }

  // ---- feats = mean over heads ----
  for (int e = tid; e < NN * NFG; e += 256) {
    const int n = e >> 3, o = e & 7;
    out_feats[((size_t)b * NN + n) * NFG + o] = facc[n][o] * 0.25f;
  }
}

// ---------------------------------------------------------------------------
extern "C" void kernel_launch(void* const* d_in, const int* in_sizes, int n_in,
                              void* d_out, int out_size, void* d_ws, size_t ws_size,
                              hipStream_t stream) {
  const float* x     = (const float*)d_in[0];
  const float* boxes = (const float*)d_in[1];
  const float* We    = (const float*)d_in[2];
  const float* be    = (const float*)d_in[3];
  const float* Wt    = (const float*)d_in[4];
  const float* bt    = (const float*)d_in[5];
  const float* Wp    = (const float*)d_in[6];
  const float* bp    = (const float*)d_in[7];
  const float* Wg    = (const float*)d_in[8];
  const int*   gt    = (const int*)d_in[9];

  float* out_feats = (float*)d_out;                       // B*N*NFG
  float* out_attn  = out_feats + (size_t)BB * NN * NFG;   // B*N*N
  float* hw        = (float*)d_ws;                        // 324 floats

  precompute_heads<<<NG, 64, 0, stream>>>(Wt, bt, Wp, bp, hw);
  gcn_fused<<<BB, 256, 0, stream>>>(x, boxes, We, be, Wg, gt, hw,
                                    out_feats, out_attn);
}